// SortMPNN_85624468013519
// MI455X (gfx1250) — compile-verified
//
#include <hip/hip_runtime.h>

#define N_NODES 8192
#define DEG     24
#define EMBED   256
#define IN_DIM  128
#define MAX_NB  32
#define G_GRAPHS 16
#define MAX_N   512
#define OUT_DIM 64

typedef __attribute__((ext_vector_type(2))) float v2f;
typedef __attribute__((ext_vector_type(8))) float v8f;

// ---------------------------------------------------------------------------
// Y[M][Nw] = X[M][K] @ W[Nw][K]^T  with fp32 WMMA 16x16x4 (exact-fp32 path).
//
// Block = 256 threads = 8 waves. The block's 64-column W strip (64 x K floats,
// <= 64KB, fits in the 320KB WGP LDS) is staged in LDS ONCE per block with
// coalesced b128 loads (8x less L2 weight traffic than per-wave fetch).
// Each wave then owns one 16-row M tile x the 64-col strip (4 accumulators);
// B fragments come from LDS (short-latency ds_load_b64) and the A fragment
// for step k+1 is prefetched from global while step k's 4 WMMAs execute.
//
// Fragment layouts (ISA 7.12.2, 32-bit):
//   A 16x4 : lane<16 -> M=lane, K=k0,k0+1 ; lane>=16 -> M=lane-16, K=k0+2,k0+3
//   B 4x16 : mirrored; per lane (n=lane&15): W[n][k0+kk], W[n][k0+kk+1]
//   C tile : VGPR r -> (M = r + 8*(lane>=16), N = lane&15)
// ---------------------------------------------------------------------------
__global__ void __launch_bounds__(256)
gemm_xwt_wmma(const float* __restrict__ X, const float* __restrict__ W,
              float* __restrict__ Y, int M, int K, int Nw) {
  __shared__ float sW[64 * 256];         // 64KB, covers K up to 256

  const int nb   = blockIdx.x;           // 64-col strip index
  const int wave = threadIdx.x >> 5;
  const int lane = threadIdx.x & 31;
  const int mn   = lane & 15;
  const int kk   = (lane >> 4) << 1;     // 0 or 2

  // ---- stage W strip into LDS (contiguous 64*K floats), coalesced b128 ----
  {
    const float4* gsrc = (const float4*)(W + (size_t)nb * 64 * K);
    float4* ldst = (float4*)sW;
    const int n4 = (64 * K) >> 2;
    for (int i = threadIdx.x; i < n4; i += 256) ldst[i] = gsrc[i];
  }
  __syncthreads();

  // ---- per-wave 16(M) x 64(N) tile ----
  const int mt = blockIdx.y * 8 + wave;
  const float* xrow = X + (size_t)(mt * 16 + mn) * K;

  const float* bs0 = sW + (size_t)( 0 + mn) * K + kk;
  const float* bs1 = sW + (size_t)(16 + mn) * K + kk;
  const float* bs2 = sW + (size_t)(32 + mn) * K + kk;
  const float* bs3 = sW + (size_t)(48 + mn) * K + kk;

  v8f c0 = {}, c1 = {}, c2 = {}, c3 = {};

  v2f a_cur = *(const v2f*)(xrow + kk);          // prefetch k-step 0
  int k0 = 0;
#pragma unroll 4
  for (; k0 < K - 4; k0 += 4) {
    v2f a_nxt = *(const v2f*)(xrow + k0 + 4 + kk);   // prefetch next step
    v2f b0 = *(const v2f*)(bs0 + k0);
    v2f b1 = *(const v2f*)(bs1 + k0);
    v2f b2 = *(const v2f*)(bs2 + k0);
    v2f b3 = *(const v2f*)(bs3 + k0);
    c0 = __builtin_amdgcn_wmma_f32_16x16x4_f32(false, a_cur, false, b0, (short)0, c0, false, false);
    c1 = __builtin_amdgcn_wmma_f32_16x16x4_f32(false, a_cur, false, b1, (short)0, c1, false, false);
    c2 = __builtin_amdgcn_wmma_f32_16x16x4_f32(false, a_cur, false, b2, (short)0, c2, false, false);
    c3 = __builtin_amdgcn_wmma_f32_16x16x4_f32(false, a_cur, false, b3, (short)0, c3, false, false);
    a_cur = a_nxt;
  }
  { // epilogue k-step (k0 == K-4)
    v2f b0 = *(const v2f*)(bs0 + k0);
    v2f b1 = *(const v2f*)(bs1 + k0);
    v2f b2 = *(const v2f*)(bs2 + k0);
    v2f b3 = *(const v2f*)(bs3 + k0);
    c0 = __builtin_amdgcn_wmma_f32_16x16x4_f32(false, a_cur, false, b0, (short)0, c0, false, false);
    c1 = __builtin_amdgcn_wmma_f32_16x16x4_f32(false, a_cur, false, b1, (short)0, c1, false, false);
    c2 = __builtin_amdgcn_wmma_f32_16x16x4_f32(false, a_cur, false, b2, (short)0, c2, false, false);
    c3 = __builtin_amdgcn_wmma_f32_16x16x4_f32(false, a_cur, false, b3, (short)0, c3, false, false);
  }

  // ---- store C tiles ----
  const int n     = lane & 15;
  const int mbase = (lane >> 4) * 8;
  float* ybase = Y + (size_t)(mt * 16 + mbase) * Nw + nb * 64 + n;
#pragma unroll
  for (int r = 0; r < 8; ++r) {
    ybase[(size_t)r * Nw +  0] = c0[r];
    ybase[(size_t)r * Nw + 16] = c1[r];
    ybase[(size_t)r * Nw + 32] = c2[r];
    ybase[(size_t)r * Nw + 48] = c3[r];
  }
}

// ---------------------------------------------------------------------------
// Fully unrolled in-register 32-element bitonic sort (ascending, matches
// jnp.sort). All indices compile-time -> pure v_min/v_max network.
// ---------------------------------------------------------------------------
__device__ __forceinline__ void bitonic_sort32(float v[MAX_NB]) {
#pragma unroll
  for (int k = 2; k <= MAX_NB; k <<= 1) {
#pragma unroll
    for (int j = k >> 1; j > 0; j >>= 1) {
#pragma unroll
      for (int i = 0; i < MAX_NB; ++i) {
        const int l = i ^ j;
        if (l > i) {
          const bool up = ((i & k) == 0);
          const float a = v[i], b = v[l];
          const float lo = fminf(a, b), hi = fmaxf(a, b);
          v[i] = up ? lo : hi;
          v[l] = up ? hi : lo;
        }
      }
    }
  }
}

// ---------------------------------------------------------------------------
// Per-node: gather the 24 projected messages (blank pad == 0, proven), sort
// the 32 slots per channel, collapse with Wc, add the combine term, write h.
// One block per node, one thread per channel -> gathers coalesced over d.
// h may alias xc (each element is read only by its own writer thread).
// ---------------------------------------------------------------------------
__global__ void __launch_bounds__(EMBED)
sort_aggregate_combine(const float* __restrict__ xp,
                       const float* xc,
                       const int* __restrict__ src,
                       const float* __restrict__ Wc,
                       float* h) {
  __shared__ int sidx[DEG];
  const int n = blockIdx.x;
  const int d = threadIdx.x;
  if (d < DEG) sidx[d] = src[n * DEG + d];
  __syncthreads();

  float v[MAX_NB];
#pragma unroll
  for (int j = 0; j < DEG; ++j)
    v[j] = xp[(size_t)sidx[j] * EMBED + d];
#pragma unroll
  for (int j = DEG; j < MAX_NB; ++j) v[j] = 0.0f;

  bitonic_sort32(v);

  float acc = xc[(size_t)n * EMBED + d];
#pragma unroll
  for (int k = 0; k < MAX_NB; ++k)
    acc = fmaf(v[k], Wc[k * EMBED + d], acc);
  h[(size_t)n * EMBED + d] = acc;
}

// ---------------------------------------------------------------------------
// Global readout: per (graph, channel) sort 512 node values in LDS (bitonic),
// dot with the Wcg column, block-reduce. Every graph has exactly MAX_N nodes,
// so no padding (blank is zero anyway).
// ---------------------------------------------------------------------------
__global__ void __launch_bounds__(256)
global_sort_collapse(const float* __restrict__ hp, const float* __restrict__ Wcg,
                     float* __restrict__ gout) {
  __shared__ float s[MAX_N];
  __shared__ float red[256];
  const int g = blockIdx.x >> 8;    // / EMBED
  const int d = blockIdx.x & 255;   // % EMBED
  const int t = threadIdx.x;

  s[t]       = hp[(size_t)(g * MAX_N + t)       * EMBED + d];
  s[t + 256] = hp[(size_t)(g * MAX_N + t + 256) * EMBED + d];
  __syncthreads();

  for (int k = 2; k <= MAX_N; k <<= 1) {
    for (int j = k >> 1; j > 0; j >>= 1) {
      const int i = ((t & ~(j - 1)) << 1) | (t & (j - 1));
      const int l = i | j;
      const bool up = ((i & k) == 0);
      const float a = s[i], b = s[l];
      const float lo = fminf(a, b), hi = fmaxf(a, b);
      s[i] = up ? lo : hi;
      s[l] = up ? hi : lo;
      __syncthreads();
    }
  }

  red[t] = fmaf(s[t], Wcg[t * EMBED + d], s[t + 256] * Wcg[(t + 256) * EMBED + d]);
  __syncthreads();
  for (int off = 128; off > 0; off >>= 1) {
    if (t < off) red[t] += red[t + off];
    __syncthreads();
  }
  if (t == 0) gout[g * EMBED + d] = red[0];
}

// ---------------------------------------------------------------------------
// out[g][o] = sum_d gsum[g][d] * Wout[o][d]   (16x64, trivial size)
// ---------------------------------------------------------------------------
__global__ void __launch_bounds__(256)
out_linear(const float* __restrict__ gsum, const float* __restrict__ Wout,
           float* __restrict__ out) {
  const int tid = blockIdx.x * 256 + threadIdx.x;   // 0..1023
  const int g = tid >> 6, o = tid & 63;
  float acc = 0.0f;
  for (int d = 0; d < EMBED; ++d)
    acc = fmaf(gsum[g * EMBED + d], Wout[o * EMBED + d], acc);
  out[tid] = acc;
}

extern "C" void kernel_launch(void* const* d_in, const int* in_sizes, int n_in,
                              void* d_out, int out_size, void* d_ws, size_t ws_size,
                              hipStream_t stream) {
  (void)in_sizes; (void)n_in; (void)out_size; (void)ws_size;
  const float* x      = (const float*)d_in[0];
  const int*   eidx   = (const int*)  d_in[1];   // [2,E] flat: first E = src
  /* d_in[2] = batch: implicit (node / 512), unused */
  const float* Wp0    = (const float*)d_in[3];
  const float* Wcomb0 = (const float*)d_in[4];
  const float* Wc0    = (const float*)d_in[5];
  const float* Wp1    = (const float*)d_in[6];
  const float* Wcomb1 = (const float*)d_in[7];
  const float* Wc1    = (const float*)d_in[8];
  const float* Wpg    = (const float*)d_in[9];
  const float* Wcg    = (const float*)d_in[10];
  const float* Wout   = (const float*)d_in[11];
  const int*   src    = eidx;

  const size_t NE = (size_t)N_NODES * EMBED;
  float* A    = (float*)d_ws;        // projected buffer
  float* B    = A + NE;              // combine / h0
  float* C    = B + NE;              // combine / h1
  float* gsum = C + NE;              // [G, EMBED]

  const dim3 blk(256);
  const dim3 gGemm(EMBED / 64, N_NODES / 128);   // (4, 64) blocks of 8 waves

  // Layer 0 (K = IN_DIM)
  gemm_xwt_wmma<<<gGemm, blk, 0, stream>>>(x, Wp0,    A, N_NODES, IN_DIM, EMBED);
  gemm_xwt_wmma<<<gGemm, blk, 0, stream>>>(x, Wcomb0, B, N_NODES, IN_DIM, EMBED);
  sort_aggregate_combine<<<N_NODES, EMBED, 0, stream>>>(A, B, src, Wc0, B);  // h0 -> B

  // Layer 1 (K = EMBED)
  gemm_xwt_wmma<<<gGemm, blk, 0, stream>>>(B, Wp1,    A, N_NODES, EMBED, EMBED);
  gemm_xwt_wmma<<<gGemm, blk, 0, stream>>>(B, Wcomb1, C, N_NODES, EMBED, EMBED);
  sort_aggregate_combine<<<N_NODES, EMBED, 0, stream>>>(A, C, src, Wc1, C);  // h1 -> C

  // Global readout
  gemm_xwt_wmma<<<gGemm, blk, 0, stream>>>(C, Wpg, A, N_NODES, EMBED, EMBED);
  global_sort_collapse<<<G_GRAPHS * EMBED, 256, 0, stream>>>(A, Wcg, gsum);
  out_linear<<<(G_GRAPHS * OUT_DIM) / 256, 256, 0, stream>>>(gsum, Wout, (float*)d_out);
}